// EdgeConv_3212635538106
// MI455X (gfx1250) — compile-verified
//
#include <hip/hip_runtime.h>
#include <hip/hip_bf16.h>

// EdgeConv (N=8192 nodes, K=64 const degree, d=128) optimized for MI455X/gfx1250.
//
// Algebra: e(i,c) = relu(concat(x_i, x_c - x_i) @ W + b)
//                 = relu( x_i @ (W1-W2) + x_c @ W2 + b )  with W = [W1;W2]
// => precompute U = X@(W1-W2)+b and V = X@W2 (two tiny fp32 WMMA GEMMs),
//    then the edge pass is add+relu+norm^2 with an argmax kept in registers.
// U,V are 4MB each -> L2-resident; only the one-time adjacency scan hits HBM.

typedef __attribute__((ext_vector_type(2))) float v2f;
typedef __attribute__((ext_vector_type(8))) float v8f;

#define DIM   128          // d_in == d_out == 128 (reference)
#define KDEG  64           // constant node degree (reference: k = 64)

// ---------------------------------------------------------------- Wd = W1-W2
__global__ void prep_wd_kernel(const float* __restrict__ W, float* __restrict__ Wd) {
    int idx = blockIdx.x * blockDim.x + threadIdx.x;   // 128*128 elements
    Wd[idx] = W[idx] - W[idx + DIM * DIM];
}

// ------------------------------------------------ adjacency -> neighbor lists
// One wave32 per row; ballot+popc compaction. Matches jnp.nonzero(A - I)
// row-major order (columns ascending within a row). Assumes degree == KDEG.
__global__ void extract_nbrs_kernel(const int* __restrict__ A,
                                    int* __restrict__ nbr, int N) {
    const int lane = threadIdx.x & 31;
    const int wave = threadIdx.x >> 5;
    const int row  = blockIdx.x * 8 + wave;
    if (row >= N) return;
    const int* Arow = A + (long long)row * N;
    int cnt = 0;
    for (int c0 = 0; c0 < N; c0 += 32) {
        int c = c0 + lane;
        bool p = (Arow[c] == 1) && (c != row);          // strip self-loop
        unsigned mask = __builtin_amdgcn_ballot_w32(p); // wave32 ballot
        if (p) {
            int off = __popc(mask & ((1u << lane) - 1u));
            int slot = cnt + off;
            if (slot < KDEG) nbr[row * KDEG + slot] = c;
        }
        cnt += __popc(mask);
    }
}

// ------------------------------------------------------------- U/V fp32 GEMM
// grid = (N/128, DIM/16, 2); block = 256 (8 waves). Each wave: one 16x16 tile.
// z==0: Out=U with Wm=Wd and +bias ; z==1: Out=V with Wm=W2.
// V_WMMA_F32_16X16X4_F32, 32 k-steps. Fragment layouts per CDNA5 ISA 7.12.2:
//   A 16x4: lanes 0-15 hold (M=lane, K=0 in v0, K=1 in v1); lanes 16-31 K=2/3.
//   B 4x16: lanes 0-15 hold (N=lane, K=0 in v0, K=1 in v1); lanes 16-31 K=2/3.
//   C/D 16x16: vgpr r -> (M = r + 8*(lane>=16), N = lane%16).
__global__ void gemm_uv_kernel(const float* __restrict__ X,
                               const float* __restrict__ Wd,
                               const float* __restrict__ W2,
                               const float* __restrict__ bias,
                               float* __restrict__ U,
                               float* __restrict__ V) {
    const int lane = threadIdx.x & 31;
    const int wave = threadIdx.x >> 5;
    const int row0 = (blockIdx.x * 8 + wave) * 16;
    const int col0 = blockIdx.y * 16;
    const bool isU = (blockIdx.z == 0);
    const float* __restrict__ Wm = isU ? Wd : W2;
    float* __restrict__ Out      = isU ? U  : V;

    const int half = lane >> 4;        // 0: K 0/1 ; 1: K 2/3
    const int l16  = lane & 15;
    const float* __restrict__ Xrow = X + (long long)(row0 + l16) * DIM;

    v8f c = {};
    for (int k0 = 0; k0 < DIM; k0 += 4) {
        const int ka = k0 + 2 * half;
        v2f a, b;
        a.x = Xrow[ka + 0];
        a.y = Xrow[ka + 1];
        b.x = Wm[(ka + 0) * DIM + col0 + l16];
        b.y = Wm[(ka + 1) * DIM + col0 + l16];
        // 8 args: (neg_a, A, neg_b, B, c_mod, C, reuse_a, reuse_b)
        c = __builtin_amdgcn_wmma_f32_16x16x4_f32(
                false, a, false, b, (short)0, c, false, false);
    }

    const float bb = isU ? bias[col0 + l16] : 0.0f;
#pragma unroll
    for (int r = 0; r < 8; ++r) {
        const int m = r + 8 * half;
        Out[(long long)(row0 + m) * DIM + col0 + l16] = c[r] + bb;
    }
}

// ---------------------------------------------------------------- edge pass
// One wave per node; lane owns channels [4*lane, 4*lane+3].
// Per edge: t = relu(u+v); norm2 via 4 FMA + 5-step shfl_xor butterfly;
// best edge's feature tracked in registers (no recompute pass).
__global__ void edge_argmax_kernel(const float* __restrict__ U,
                                   const float* __restrict__ V,
                                   const int* __restrict__ nbr,
                                   float* __restrict__ out, int N) {
    const int lane = threadIdx.x & 31;
    const int wave = threadIdx.x >> 5;
    const int node = blockIdx.x * 8 + wave;
    if (node >= N) return;

    const float4 u = ((const float4*)(U + (long long)node * DIM))[lane];
    const int* __restrict__ nb = nbr + node * KDEG;

    float4 best_e = make_float4(0.f, 0.f, 0.f, 0.f);
    float  best_n = -1.0f;

    for (int j = 0; j < KDEG; ++j) {
        const int c = nb[j];                                   // wave-uniform
        const float4 v = ((const float4*)(V + (long long)c * DIM))[lane];
        float4 t;
        t.x = fmaxf(u.x + v.x, 0.0f);
        t.y = fmaxf(u.y + v.y, 0.0f);
        t.z = fmaxf(u.z + v.z, 0.0f);
        t.w = fmaxf(u.w + v.w, 0.0f);
        float s = t.x * t.x;
        s = fmaf(t.y, t.y, s);
        s = fmaf(t.z, t.z, s);
        s = fmaf(t.w, t.w, s);
        // wave32 all-reduce (sum across 32 lanes)
        s += __shfl_xor(s, 16, 32);
        s += __shfl_xor(s, 8, 32);
        s += __shfl_xor(s, 4, 32);
        s += __shfl_xor(s, 2, 32);
        s += __shfl_xor(s, 1, 32);
        if (s > best_n) { best_n = s; best_e = t; }
    }
    ((float4*)(out + (long long)node * DIM))[lane] = best_e;
}

// ----------------------------------------------------------------- launcher
extern "C" void kernel_launch(void* const* d_in, const int* in_sizes, int n_in,
                              void* d_out, int out_size, void* d_ws, size_t ws_size,
                              hipStream_t stream) {
    const int*   A = (const int*)d_in[0];     // [N, N] int32
    const float* X = (const float*)d_in[1];   // [N, 128] f32
    const float* W = (const float*)d_in[2];   // [256, 128] f32
    const float* b = (const float*)d_in[3];   // [128] f32
    // d_in[4] = k (device scalar); degree fixed at KDEG=64 per reference setup.

    const int N = in_sizes[1] / DIM;          // 8192

    // Workspace layout (256B aligned): nbr | Wd | U | V  (~10.1 MB total)
    char* ws = (char*)d_ws;
    size_t off = 0;
    int*   nbr = (int*)(ws + off);   off += (size_t)N * KDEG * sizeof(int);
    off = (off + 255) & ~(size_t)255;
    float* Wd  = (float*)(ws + off); off += (size_t)DIM * DIM * sizeof(float);
    off = (off + 255) & ~(size_t)255;
    float* Uu  = (float*)(ws + off); off += (size_t)N * DIM * sizeof(float);
    off = (off + 255) & ~(size_t)255;
    float* Vv  = (float*)(ws + off);
    const float* W2 = W + DIM * DIM;

    prep_wd_kernel<<<(DIM * DIM) / 256, 256, 0, stream>>>(W, Wd);
    extract_nbrs_kernel<<<(N + 7) / 8, 256, 0, stream>>>(A, nbr, N);
    gemm_uv_kernel<<<dim3(N / 128, DIM / 16, 2), 256, 0, stream>>>(
        X, Wd, W2, b, Uu, Vv);
    edge_argmax_kernel<<<(N + 7) / 8, 256, 0, stream>>>(
        Uu, Vv, nbr, (float*)d_out, N);
}